// NonMaxSuppression_1872605741345
// MI455X (gfx1250) — compile-verified
//
#include <hip/hip_runtime.h>
#include <stdint.h>

// NMS + ordered compaction for 4096x4096 maps (see reference).
// Memory-bound: ~140MB of mandatory traffic -> ~6us at 23.3 TB/s HBM.
// CDNA5-specific paths used:
//   * tensor_load_to_lds (TDM): ONE instruction per workgroup stages the
//     3x4096 f32 stencil tile into LDS; drained with s_wait_tensorcnt.
//   * global_load_async_to_lds_b128 (+ s_wait_asynccnt) fallback for the two
//     image-border rows (ASYNCcnt path).
//   * wave32 ballot packs the maxima mask: one ballot == one 32-bit mask word
//     in row-major x order.

#define H_DIM 4096
#define W_DIM 4096
#define MAX_K (1 << 20)
#define REL_THR 0.7f
#define REP_THR 0.7f

#define WORDS_PER_ROW (W_DIM / 32)             // 128
#define MASK_WORDS (H_DIM * WORDS_PER_ROW)     // 524288 u32 = 2 MB

typedef uint32_t v4u __attribute__((ext_vector_type(4)));
typedef int      v8i __attribute__((ext_vector_type(8)));
typedef int      v4i __attribute__((ext_vector_type(4)));

__device__ __forceinline__ void async_load_b128_to_lds(uint32_t lds_byte,
                                                       uint64_t gaddr) {
  // gfx1250 async DMA: global memory -> LDS, tracked by ASYNCcnt.
  asm volatile("global_load_async_to_lds_b128 %0, %1, off"
               :: "v"(lds_byte), "v"(gaddr)
               : "memory");
}

__device__ __forceinline__ void wait_asynccnt0() {
  asm volatile("s_wait_asynccnt 0x0" ::: "memory");
}

// TDM: load a 2D tile (tile_d0 elements wide x tile_d1 rows, f32) from a
// row-major tensor with stride `t_d0_stride` (elements) into contiguous LDS.
// Descriptor bitfields per CDNA5 ISA 8.3/8.4 (D# group0 / group1).
// This toolchain's builtin takes 6 args: (g0, g1, g2, g3, <v8i extra>, cpol).
__device__ __forceinline__ void tdm_load_2d_f32(uint32_t lds_byte,
                                                uint64_t gaddr,
                                                uint32_t tile_d0,
                                                uint32_t tile_d1,
                                                uint32_t t_d0_stride) {
  v4u g0;
  g0.x = 1u;                                   // count=1 (valid), no gather
  g0.y = lds_byte;                             // lds_addr [63:32]
  g0.z = (uint32_t)gaddr;                      // global_addr [95:64]
  g0.w = (uint32_t)((gaddr >> 32) & 0x01FFFFFFull)  // global_addr [120:96]
         | 0x80000000u;                        // type=2 ("image") [127:126]
  v8i g1;
  g1[0] = 0x00020000;                          // data_size=2 (4B) [17:16]
  g1[1] = (int)((tile_d0 & 0xFFFFu) << 16);    // tensor_dim0[15:0] -> [63:48]
  g1[2] = (int)(((tile_d0 >> 16) & 0xFFFFu)    // tensor_dim0[31:16]
         | ((tile_d1 & 0xFFFFu) << 16));       // tensor_dim1[15:0] -> [95:80]
  g1[3] = (int)(((tile_d1 >> 16) & 0xFFFFu)    // tensor_dim1[31:16]
         | ((tile_d0 & 0xFFFFu) << 16));       // tile_dim0 [127:112]
  g1[4] = (int)(tile_d1 & 0xFFFFu);            // tile_dim1 [143:128]; tile_dim2=0
  g1[5] = (int)t_d0_stride;                    // tensor_dim0_stride[31:0]
  g1[6] = 0;                                   // stride hi / dim1_stride (2D: unused)
  g1[7] = 0;
  v4i g2 = {0, 0, 0, 0};                       // 2D tile: groups 2/3 unused
  v4i g3 = {0, 0, 0, 0};
  v8i gx = {0, 0, 0, 0, 0, 0, 0, 0};           // extra operand (zero-filled)
  __builtin_amdgcn_tensor_load_to_lds(g0, g1, g2, g3, gx, 0);
}

// ---------------------------------------------------------------------------
// K0: fill output with -1 (padded slots of jnp.nonzero(..., fill_value=-1)).
// ---------------------------------------------------------------------------
__global__ __launch_bounds__(256) void nms_fill_out(int4* __restrict__ out) {
  const int i = blockIdx.x * 256 + threadIdx.x;
  out[i] = make_int4(-1, -1, -1, -1);
}

// ---------------------------------------------------------------------------
// K1: per-row maxima mask (bit-packed) + per-row count.
// One block per image row; 256 threads. Rows y-1..y+1 of `rep` staged into a
// contiguous LDS tile: interior rows via a single TDM tensor_load_to_lds,
// border rows via async-to-LDS + -inf fill (maxpool's -inf padding).
// ---------------------------------------------------------------------------
__global__ __launch_bounds__(256) void nms_mask_count(
    const float* __restrict__ rel, const float* __restrict__ rep,
    uint32_t* __restrict__ mask, uint32_t* __restrict__ counts) {
  __shared__ __align__(16) float tile[3 * W_DIM];   // 48 KB, rows contiguous
  __shared__ uint32_t s_cnt;

  const int t = threadIdx.x;
  const int y = blockIdx.x;
  const float ninf = -__builtin_inff();
  if (t == 0) s_cnt = 0;

  const uint32_t tile_lds = (uint32_t)(uintptr_t)&tile[0];
  const bool interior = (y > 0) && (y < H_DIM - 1);

  if (interior) {
    // One TDM DMA moves the whole 3x4096 tile; issue from wave 0 only
    // (TDM ignores EXEC; descriptor operands are wave-uniform SGPRs).
    if (t < 32) {
      const uint64_t gaddr =
          (uint64_t)(uintptr_t)(rep + (size_t)(y - 1) * W_DIM);
      tdm_load_2d_f32(tile_lds, gaddr, W_DIM, 3u, W_DIM);
    }
  } else {
    // Border rows: valid rows via async-to-LDS DMA, OOB row filled with -inf.
    for (int r = 0; r < 3; ++r) {
      const int ry = y - 1 + r;
      if (ry >= 0 && ry < H_DIM) {
        const uint64_t rowbase =
            (uint64_t)(uintptr_t)(rep + (size_t)ry * W_DIM);
        const uint32_t ldsrow = tile_lds + (uint32_t)r * (W_DIM * 4u);
#pragma unroll
        for (int s = 0; s < 4; ++s) {
          const uint32_t c = (uint32_t)t + 256u * s;  // 16-byte chunk index
          async_load_b128_to_lds(ldsrow + 16u * c, rowbase + 16ull * c);
        }
      } else {
        const float4 m = make_float4(ninf, ninf, ninf, ninf);
        float4* dst = (float4*)&tile[r * W_DIM];
#pragma unroll
        for (int s = 0; s < 4; ++s) dst[t + 256 * s] = m;
      }
    }
  }
  wait_asynccnt0();                       // drain ASYNCcnt (border path)
  if (t < 32) __builtin_amdgcn_s_wait_tensorcnt(0);  // drain TDM (wave 0)
  __syncthreads();                        // staged tile visible to all waves

  const int lane = t & 31;
  const int wv = t >> 5;
  const float* relRow = rel + (size_t)y * W_DIM;
  uint32_t* maskRow = mask + (size_t)y * WORDS_PER_ROW;

#pragma unroll 4
  for (int k = 0; k < 16; ++k) {
    const int x = t + 256 * k;        // lanes of a wave cover 32 consecutive x
    const bool hasL = (x > 0);
    const bool hasR = (x < W_DIM - 1);
    const float c = tile[W_DIM + x];
    float m = tile[x];                                   // up
    m = fmaxf(m, tile[2 * W_DIM + x]);                   // down
    m = fmaxf(m, hasL ? tile[x - 1] : ninf);             // up-left
    m = fmaxf(m, hasL ? tile[W_DIM + x - 1] : ninf);     // left
    m = fmaxf(m, hasL ? tile[2 * W_DIM + x - 1] : ninf); // down-left
    m = fmaxf(m, hasR ? tile[x + 1] : ninf);             // up-right
    m = fmaxf(m, hasR ? tile[W_DIM + x + 1] : ninf);     // right
    m = fmaxf(m, hasR ? tile[2 * W_DIM + x + 1] : ninf); // down-right
    // c == maxpool3x3  <=>  c >= all 8 neighbors
    const bool pred = (c >= m) && (c >= REP_THR) && (relRow[x] >= REL_THR);
    const uint32_t bal = __builtin_amdgcn_ballot_w32(pred);  // bit i == x offset i
    if (lane == 0) {
      maskRow[wv + 8 * k] = bal;  // word w covers x in [32w, 32w+32)
      if (bal) atomicAdd(&s_cnt, (uint32_t)__popc(bal));
    }
  }
  __syncthreads();
  if (t == 0) counts[y] = s_cnt;
}

// ---------------------------------------------------------------------------
// K2: single-block exclusive scan of 4096 row counts -> row base offsets.
// Preserves the row-major global ordering of jnp.nonzero.
// ---------------------------------------------------------------------------
__global__ __launch_bounds__(1024) void nms_scan_counts(
    const uint32_t* __restrict__ counts, uint32_t* __restrict__ offsets) {
  __shared__ uint32_t s[1024];
  const int t = threadIdx.x;
  const uint32_t c0 = counts[4 * t + 0];
  const uint32_t c1 = counts[4 * t + 1];
  const uint32_t c2 = counts[4 * t + 2];
  const uint32_t c3 = counts[4 * t + 3];
  const uint32_t sum = c0 + c1 + c2 + c3;
  s[t] = sum;
  __syncthreads();
  for (int off = 1; off < 1024; off <<= 1) {
    const uint32_t v = (t >= off) ? s[t - off] : 0u;
    __syncthreads();
    s[t] += v;
    __syncthreads();
  }
  const uint32_t excl = s[t] - sum;
  offsets[4 * t + 0] = excl;
  offsets[4 * t + 1] = excl + c0;
  offsets[4 * t + 2] = excl + c0 + c1;
  offsets[4 * t + 3] = excl + c0 + c1 + c2;
}

// ---------------------------------------------------------------------------
// K3: emit (y, x) pairs in row-major order. One block per row; 128 threads,
// one mask word each; small scan gives per-word exclusive rank within the row.
// ---------------------------------------------------------------------------
__global__ __launch_bounds__(128) void nms_emit(
    const uint32_t* __restrict__ mask, const uint32_t* __restrict__ offsets,
    int* __restrict__ out) {
  __shared__ uint32_t wp[128];
  const int t = threadIdx.x;
  const int y = blockIdx.x;
  uint32_t w = mask[(size_t)y * WORDS_PER_ROW + t];
  const uint32_t pc = (uint32_t)__popc(w);
  wp[t] = pc;
  __syncthreads();
  for (int off = 1; off < 128; off <<= 1) {
    const uint32_t v = (t >= off) ? wp[t - off] : 0u;
    __syncthreads();
    wp[t] += v;
    __syncthreads();
  }
  if (w) {
    uint32_t r = offsets[y] + wp[t] - pc;  // exclusive rank of this word
    const int xbase = t * 32;
    while (w) {
      const int b = __ffs(w) - 1;  // LSB-first == ascending x
      w &= w - 1;
      if (r < MAX_K) {
        out[r] = y;                  // row of y indices
        out[MAX_K + r] = xbase + b;  // row of x indices
      }
      ++r;
    }
  }
}

// ---------------------------------------------------------------------------
extern "C" void kernel_launch(void* const* d_in, const int* in_sizes, int n_in,
                              void* d_out, int out_size, void* d_ws,
                              size_t ws_size, hipStream_t stream) {
  const float* rel = (const float*)d_in[0];  // reliability
  const float* rep = (const float*)d_in[1];  // repeatability
  int* out = (int*)d_out;                    // [2, MAX_K] int32

  // Workspace layout (~2.03 MB): bitmask | per-row counts | per-row offsets
  uint32_t* mask = (uint32_t*)d_ws;
  uint32_t* counts = mask + MASK_WORDS;
  uint32_t* offsets = counts + H_DIM;

  nms_fill_out<<<(2 * MAX_K / 4) / 256, 256, 0, stream>>>((int4*)out);
  nms_mask_count<<<H_DIM, 256, 0, stream>>>(rel, rep, mask, counts);
  nms_scan_counts<<<1, 1024, 0, stream>>>(counts, offsets);
  nms_emit<<<H_DIM, 128, 0, stream>>>(mask, offsets, out);
}